// LinearTimeMMDLoss_73821897883957
// MI455X (gfx1250) — compile-verified
//
#include <hip/hip_runtime.h>
#include <hip/hip_bf16.h>

// LinearTimeMMD loss — MI455X (gfx1250, wave32).
// Roofline: 256 MB streamed once -> ~11 us at 23.3 TB/s; ~0.4 GFLOP -> purely
// HBM-bound. Strategy: wave-per-pair b128 coalesced access; CDNA5 async
// global->LDS staging (ASYNCcnt path) with per-wave double buffering when the
// toolchain exposes the gfx1250 builtins, else direct non-temporal loads.
// Final reduction: deterministic two-stage (per-block partials in d_ws, then
// one tiny reduce kernel) — no global float atomics (2048 serialized atomics
// on one address would cost on the order of the whole kernel's runtime).

typedef float v4f __attribute__((ext_vector_type(4)));
typedef int   v4i __attribute__((vector_size(16)));   // b128 payload type

#define DIM 512                       // feature dimension
#define WAVES_PER_BLOCK 4
#define PAIRS_PER_WAVE 4
#define BLOCK_THREADS (WAVES_PER_BLOCK * 32)
#define ROW_BYTES  (DIM * 4)          // 2048 B per row
#define PAIR_BYTES (4 * ROW_BYTES)    // 8192 B: x_odd, x_even, y_odd, y_even
#define WAVE_LDS   (2 * PAIR_BYTES)   // 16384 B double buffer per wave

#if defined(__AMDGCN__) && __has_builtin(__builtin_amdgcn_global_load_async_to_lds_b128)
#define USE_ASYNC 1
#else
#define USE_ASYNC 0
#endif

#if USE_ASYNC
// Builtin signature (from hipcc diagnostic): arg0 = AS1 pointer to v4i
// (global source), arg1 = LDS pointer, then imm offset + imm cpol.
typedef __attribute__((address_space(1))) v4i gas_v4i;
typedef __attribute__((address_space(3))) v4i las_v4i;

// Copy one 2 KB row: 4 x (32 lanes x 16 B) async transfers (ASYNCcnt += 4).
__device__ __forceinline__ void async_row(const float* __restrict__ g, char* l, int lane) {
#pragma unroll
  for (int c = 0; c < 4; ++c) {
    __builtin_amdgcn_global_load_async_to_lds_b128(
        (gas_v4i*)(g + c * 128 + lane * 4),
        (las_v4i*)(l + c * 512 + lane * 16),
        /*offset=*/0, /*cpol=*/0);
  }
}

// Stage one pair (8 KB, ASYNCcnt += 16).
__device__ __forceinline__ void issue_pair(const float* __restrict__ src,
                                           const float* __restrict__ tgt,
                                           int p, char* buf, int lane) {
  const float* xs = src + (size_t)p * (2 * DIM);
  const float* ys = tgt + (size_t)p * (2 * DIM);
  async_row(xs,       buf + 0 * ROW_BYTES, lane);   // x_odd
  async_row(xs + DIM, buf + 1 * ROW_BYTES, lane);   // x_even
  async_row(ys,       buf + 2 * ROW_BYTES, lane);   // y_odd
  async_row(ys + DIM, buf + 3 * ROW_BYTES, lane);   // y_even
}

__device__ __forceinline__ void async_wait_le16() {
#if __has_builtin(__builtin_amdgcn_s_wait_asynccnt)
  __builtin_amdgcn_s_wait_asynccnt(16);
#else
  asm volatile("s_wait_asynccnt 0x10" ::: "memory");
#endif
}
__device__ __forceinline__ void async_wait_le0() {
#if __has_builtin(__builtin_amdgcn_s_wait_asynccnt)
  __builtin_amdgcn_s_wait_asynccnt(0);
#else
  asm volatile("s_wait_asynccnt 0x0" ::: "memory");
#endif
}
__device__ __forceinline__ void ds_drain() {
  // Belt-and-braces: make sure our LDS reads retired before the freed buffer
  // is overwritten by the next async stage (async LDS writes are unordered
  // w.r.t. this wave's DS ops).
  asm volatile("s_wait_dscnt 0x0" ::: "memory");
}
#endif  // USE_ASYNC

// Accumulate the four squared distances for one pair; rows at xo/xe/yo/ye.
__device__ __forceinline__ void accum_pair(const float* xo, const float* xe,
                                           const float* yo, const float* ye,
                                           int lane, float& dxx, float& dyy,
                                           float& dxy, float& dyx) {
  dxx = dyy = dxy = dyx = 0.0f;
#pragma unroll
  for (int it = 0; it < DIM / (32 * 4); ++it) {   // 4 iterations
    const int off = it * 128 + lane * 4;
    v4f a = *(const v4f*)(xo + off);
    v4f b = *(const v4f*)(xe + off);
    v4f c = *(const v4f*)(yo + off);
    v4f d = *(const v4f*)(ye + off);
#pragma unroll
    for (int k = 0; k < 4; ++k) {
      float t0 = a[k] - b[k]; dxx = fmaf(t0, t0, dxx);
      float t1 = c[k] - d[k]; dyy = fmaf(t1, t1, dyy);
      float t2 = a[k] - d[k]; dxy = fmaf(t2, t2, dxy);
      float t3 = b[k] - c[k]; dyx = fmaf(t3, t3, dyx);
    }
  }
}

// Wave butterfly + 5-bandwidth Gaussian epilogue (uniform across lanes).
__device__ __forceinline__ float pair_epilogue(float dxx, float dyy,
                                               float dxy, float dyx) {
#pragma unroll
  for (int m = 16; m >= 1; m >>= 1) {
    dxx += __shfl_xor(dxx, m, 32);
    dyy += __shfl_xor(dyy, m, 32);
    dxy += __shfl_xor(dxy, m, 32);
    dyx += __shfl_xor(dyx, m, 32);
  }
  // bandwidth = (sum)/4 / 2^(5//2) = sum * 0.0625 ; bws[i] = bw * 2^i
  float bw  = (dxx + dyy + dxy + dyx) * 0.0625f;
  float inv = 1.0f / bw;
  float h = 0.0f;
#pragma unroll
  for (int i = 0; i < 5; ++i) {
    h += expf(-dxx * inv) + expf(-dyy * inv)
       - expf(-dxy * inv) - expf(-dyx * inv);
    inv *= 0.5f;
  }
  return h;
}

__global__ __launch_bounds__(BLOCK_THREADS)
void mmd_main_kernel(const float* __restrict__ src,
                     const float* __restrict__ tgt,
                     float* __restrict__ partial, int m2) {
  constexpr int SLOT_STRIDE = USE_ASYNC ? WAVE_LDS : 16;
  __shared__ __align__(16) char smem[WAVES_PER_BLOCK * SLOT_STRIDE];  // 64 KB async / 64 B fallback

  const int lane = threadIdx.x & 31;   // wave32
  const int wave = threadIdx.x >> 5;
  char* buf = smem + wave * SLOT_STRIDE;
  const int base = (blockIdx.x * WAVES_PER_BLOCK + wave) * PAIRS_PER_WAVE;

  float h_acc = 0.0f;

  if (base + PAIRS_PER_WAVE <= m2) {
#if USE_ASYNC
    // Software pipeline: p0,p1 staged up front; steady state overlaps the
    // global->LDS DMA of pair i+2 with the compute of pair i.
    issue_pair(src, tgt, base + 0, buf, lane);               // ASYNCcnt 16
    issue_pair(src, tgt, base + 1, buf + PAIR_BYTES, lane);  // ASYNCcnt 32
#pragma unroll
    for (int i = 0; i < PAIRS_PER_WAVE; ++i) {
      if (i < PAIRS_PER_WAVE - 1) async_wait_le16();  // pair i landed
      else                        async_wait_le0();   // last pair landed
      char* b = buf + (i & 1) * PAIR_BYTES;
      float dxx, dyy, dxy, dyx;
      accum_pair((const float*)(b + 0 * ROW_BYTES), (const float*)(b + 1 * ROW_BYTES),
                 (const float*)(b + 2 * ROW_BYTES), (const float*)(b + 3 * ROW_BYTES),
                 lane, dxx, dyy, dxy, dyx);
      if (i + 2 < PAIRS_PER_WAVE) {
        ds_drain();                                   // LDS reads retired
        issue_pair(src, tgt, base + i + 2, b, lane);  // refill freed buffer
      }
      h_acc += pair_epilogue(dxx, dyy, dxy, dyx);
    }
#else
    // Fallback: direct coalesced non-temporal b128 loads (streamed once;
    // keep the 192 MB L2 from being churned), gfx1250 prefetch of the
    // wave's upcoming 4 KB regions.
#pragma unroll 1
    for (int i = 0; i < PAIRS_PER_WAVE; ++i) {
      const int p = base + i;
      const float* xs = src + (size_t)p * (2 * DIM);
      const float* ys = tgt + (size_t)p * (2 * DIM);
      __builtin_prefetch(xs + lane * 32, 0, 0);
      __builtin_prefetch(ys + lane * 32, 0, 0);
      float dxx = 0.0f, dyy = 0.0f, dxy = 0.0f, dyx = 0.0f;
#pragma unroll
      for (int it = 0; it < DIM / (32 * 4); ++it) {
        const int off = it * 128 + lane * 4;
        v4f a = __builtin_nontemporal_load((const v4f*)(xs + off));
        v4f b = __builtin_nontemporal_load((const v4f*)(xs + DIM + off));
        v4f c = __builtin_nontemporal_load((const v4f*)(ys + off));
        v4f d = __builtin_nontemporal_load((const v4f*)(ys + DIM + off));
#pragma unroll
        for (int k = 0; k < 4; ++k) {
          float t0 = a[k] - b[k]; dxx = fmaf(t0, t0, dxx);
          float t1 = c[k] - d[k]; dyy = fmaf(t1, t1, dyy);
          float t2 = a[k] - d[k]; dxy = fmaf(t2, t2, dxy);
          float t3 = b[k] - c[k]; dyx = fmaf(t3, t3, dyx);
        }
      }
      h_acc += pair_epilogue(dxx, dyy, dxy, dyx);
    }
#endif
  }

  // Per-wave result -> this wave's own LDS slot (its buffers are done).
  if (lane == 0) *(float*)buf = h_acc;
  __syncthreads();
  if (threadIdx.x == 0) {
    float s = 0.0f;
#pragma unroll
    for (int w = 0; w < WAVES_PER_BLOCK; ++w)
      s += *(const float*)(smem + w * SLOT_STRIDE);
    partial[blockIdx.x] = s;           // deterministic per-block partial
  }
}

#define REDUCE_THREADS 256
__global__ __launch_bounds__(REDUCE_THREADS)
void mmd_reduce_kernel(const float* __restrict__ partial, int n,
                       float inv_m2, float* __restrict__ out) {
  __shared__ float wsum[REDUCE_THREADS / 32];
  float a = 0.0f;
  for (int i = threadIdx.x; i < n; i += REDUCE_THREADS) a += partial[i];
#pragma unroll
  for (int m = 16; m >= 1; m >>= 1) a += __shfl_xor(a, m, 32);
  if ((threadIdx.x & 31) == 0) wsum[threadIdx.x >> 5] = a;
  __syncthreads();
  if (threadIdx.x == 0) {
    float s = 0.0f;
#pragma unroll
    for (int w = 0; w < REDUCE_THREADS / 32; ++w) s += wsum[w];
    *out = s * inv_m2;
  }
}

extern "C" void kernel_launch(void* const* d_in, const int* in_sizes, int n_in,
                              void* d_out, int out_size, void* d_ws, size_t ws_size,
                              hipStream_t stream) {
  const float* src = (const float*)d_in[0];
  const float* tgt = (const float*)d_in[1];
  float* out = (float*)d_out;
  float* partial = (float*)d_ws;       // one float per block (8 KB total)

  const int m  = in_sizes[0] / DIM;    // 65536 rows
  const int m2 = m / 2;                // 32768 pairs

  const int pairs_per_block = WAVES_PER_BLOCK * PAIRS_PER_WAVE;  // 16
  const int blocks = (m2 + pairs_per_block - 1) / pairs_per_block;  // 2048

  mmd_main_kernel<<<blocks, BLOCK_THREADS, 0, stream>>>(src, tgt, partial, m2);
  mmd_reduce_kernel<<<1, REDUCE_THREADS, 0, stream>>>(partial, blocks,
                                                      1.0f / (float)m2, out);
}